// Chamferloss_48292612276314
// MI455X (gfx1250) — compile-verified
//
#include <hip/hip_runtime.h>
#include <hip/hip_bf16.h>

// Chamfer loss, B=2, H=W=96, C=3  ->  N=9216 points per batch.
// P[i,j] = xx_i + yy_j - 2*x_i.y_j.  Using V_WMMA_F32_16X16X4_F32 with the
// K=4 pad slot carrying the norm term:
//   A row m  = (-2*x0, -2*x1, -2*x2, 1)
//   B col n  = ( y0,    y1,    y2,   yy_n)
// so the WMMA output directly equals  yy_j - 2*x_i.y_j  and the loop body is
// just load + wmma + min.  min_j P[i,j] = xx_i + min_j(wmma out); the second
// Chamfer direction reuses the same kernel with pred/label swapped.
// Each wave owns TWO 16-query tiles (two A operands) and shares every
// B-operand load between two WMMAs.

typedef __attribute__((ext_vector_type(2))) float v2f;
typedef __attribute__((ext_vector_type(8))) float v8f;

constexpr int BATCH = 2;
constexpr int NPTS  = 9216;        // 96*96
constexpr int NTIL  = NPTS / 16;   // 576 target tiles of 16 points
constexpr int QPAIR = NTIL / 2;    // 288 query-tile-pairs per (dir, b)
constexpr int WAVES_PER_BLOCK = 8;
constexpr int TASKS = 2 * BATCH * QPAIR;               // 1152 wave tasks
constexpr int MAIN_BLOCKS = TASKS / WAVES_PER_BLOCK;   // 144

// ---------------------------------------------------------------- sq norms
__global__ void chamfer_sqnorm(const float* __restrict__ pred,
                               const float* __restrict__ label,
                               float* __restrict__ sqp,
                               float* __restrict__ sql) {
  int i = blockIdx.x * blockDim.x + threadIdx.x;
  if (i < BATCH * NPTS) {
    const float* p = pred + (size_t)i * 3;
    const float* l = label + (size_t)i * 3;
    sqp[i] = p[0] * p[0] + p[1] * p[1] + p[2] * p[2];
    sql[i] = l[0] * l[0] + l[1] * l[1] + l[2] * l[2];
  }
}

// ------------------------------------------------- main WMMA nearest-neighbor
__global__ void __launch_bounds__(256)
chamfer_main(const float* __restrict__ pred, const float* __restrict__ label,
             const float* __restrict__ sqp,  const float* __restrict__ sql,
             float* __restrict__ partials) {
  const int lane = threadIdx.x & 31;
  const int wave = threadIdx.x >> 5;
  const int task = blockIdx.x * WAVES_PER_BLOCK + wave;   // [0, TASKS)

  const int dir = task / (BATCH * QPAIR);                 // 0: q=pred, 1: q=label
  const int rem = task - dir * (BATCH * QPAIR);
  const int b   = rem / QPAIR;
  const int ip  = rem - b * QPAIR;                        // query-pair index
  const int it0 = ip * 2;                                 // first of two query tiles

  const float* q  = (dir ? label : pred) + (size_t)b * NPTS * 3;
  const float* t  = (dir ? pred : label) + (size_t)b * NPTS * 3;
  const float* qn = (dir ? sql : sqp) + (size_t)b * NPTS;
  const float* tn = (dir ? sqp : sql) + (size_t)b * NPTS;

  const int  l16 = lane & 15;
  const bool lo  = lane < 16;

  // A operands (16x4 f32): lanes 0-15 hold K=0,1 = {-2*c0,-2*c1} of row m=lane;
  // lanes 16-31 hold K=2,3 = {-2*c2, 1.0} of row m=lane-16.
  const int qi0 = it0 * 16 + l16;
  const int qi1 = qi0 + 16;
  v2f a0, a1;
  {
    int off = qi0 * 3 + (lo ? 0 : 1);
    float x0 = q[off], x1 = q[off + 1];
    a0.x = -2.0f * (lo ? x0 : x1);
    a0.y = lo ? (-2.0f * x1) : 1.0f;
    off = qi1 * 3 + (lo ? 0 : 1);
    x0 = q[off]; x1 = q[off + 1];
    a1.x = -2.0f * (lo ? x0 : x1);
    a1.y = lo ? (-2.0f * x1) : 1.0f;
  }
  const float xx0 = qn[qi0];
  const float xx1 = qn[qi1];

  const v8f zero = {};
  v8f rmin0, rmin1;
#pragma unroll
  for (int r = 0; r < 8; ++r) { rmin0[r] = 3.402823466e38f; rmin1[r] = 3.402823466e38f; }

#pragma unroll 2
  for (int jt = 0; jt < NTIL; ++jt) {
    const int jn  = jt * 16 + l16;
    const int off = jn * 3 + (lo ? 0 : 1);
    const float y0  = t[off], y1 = t[off + 1];
    const float yyv = tn[jn];                  // |y_{j}|^2
    v2f bv;                                    // B: lanes<16 {y0,y1}; lanes>=16 {y2,yy}
    bv.x = lo ? y0 : y1;
    bv.y = lo ? y1 : yyv;

    // d[r] = yy_j - 2*x_i.y_j for rows m=r / m=r+8, col n=l16
    v8f d0 = __builtin_amdgcn_wmma_f32_16x16x4_f32(
        false, a0, false, bv, (short)0, zero, false, false);
    v8f d1 = __builtin_amdgcn_wmma_f32_16x16x4_f32(
        false, a1, false, bv, (short)0, zero, false, false);

#pragma unroll
    for (int r = 0; r < 8; ++r) {
      rmin0[r] = fminf(rmin0[r], d0[r]);
      rmin1[r] = fminf(rmin1[r], d1[r]);
    }
  }

  // Min across the 16 lanes sharing each row (butterfly stays within halves).
#pragma unroll
  for (int r = 0; r < 8; ++r) {
    float v0 = rmin0[r], v1 = rmin1[r];
#pragma unroll
    for (int m = 1; m < 16; m <<= 1) {
      v0 = fminf(v0, __shfl_xor(v0, m, 32));
      v1 = fminf(v1, __shfl_xor(v1, m, 32));
    }
    rmin0[r] = v0; rmin1[r] = v1;
  }

  // Add xx per row and sum the 32 rows handled by this wave.
  float S = 0.0f;
#pragma unroll
  for (int r = 0; r < 8; ++r) {
    const int src = lo ? r : (8 + r);          // lane holding xx of row m
    S += rmin0[r] + __shfl(xx0, src, 32);
    S += rmin1[r] + __shfl(xx1, src, 32);
  }
  const float waveSum = __shfl(S, 0, 32) + __shfl(S, 16, 32);

  __shared__ float wsum[WAVES_PER_BLOCK];
  if (lane == 0) wsum[wave] = waveSum;
  __syncthreads();
  if (threadIdx.x == 0) {
    float acc = 0.0f;
#pragma unroll
    for (int w = 0; w < WAVES_PER_BLOCK; ++w) acc += wsum[w];
    partials[blockIdx.x] = acc;
  }
}

// ---------------------------------------------------------------- finalize
__global__ void chamfer_finalize(const float* __restrict__ partials,
                                 float* __restrict__ out) {
  __shared__ float s[512];
  const int t = threadIdx.x;
  s[t] = (t < MAIN_BLOCKS) ? partials[t] : 0.0f;
  __syncthreads();
  for (int st = 256; st > 0; st >>= 1) {
    if (t < st) s[t] += s[t + st];
    __syncthreads();
  }
  if (t == 0) out[0] = s[0] * (1.0f / (float)BATCH);
}

extern "C" void kernel_launch(void* const* d_in, const int* in_sizes, int n_in,
                              void* d_out, int out_size, void* d_ws, size_t ws_size,
                              hipStream_t stream) {
  const float* pred  = (const float*)d_in[0];
  const float* label = (const float*)d_in[1];
  float* ws  = (float*)d_ws;
  float* sqp = ws;                       // BATCH*NPTS floats
  float* sql = ws + BATCH * NPTS;        // BATCH*NPTS floats
  float* prt = ws + 2 * BATCH * NPTS;    // MAIN_BLOCKS floats
  float* out = (float*)d_out;

  chamfer_sqnorm<<<(BATCH * NPTS + 255) / 256, 256, 0, stream>>>(pred, label, sqp, sql);
  chamfer_main<<<MAIN_BLOCKS, 256, 0, stream>>>(pred, label, sqp, sql, prt);
  chamfer_finalize<<<1, 512, 0, stream>>>(prt, out);
}